// MultiHeadAttention_21517786153329
// MI455X (gfx1250) — compile-verified
//
#include <hip/hip_runtime.h>
#include <hip/hip_bf16.h>

// MI455X / gfx1250, wave32. Full fp32 path via V_WMMA_F32_16X16X4_F32.

typedef float v2f __attribute__((ext_vector_type(2)));
typedef float v8f __attribute__((ext_vector_type(8)));

#define HID   128      // hidden size H
#define SEQ   64       // L (both peptide and cdr3)
#define NBATCH 256
#define LDSP  132      // padded LDS row stride in floats (multiple of 4 for float4 stores,
                       // 132 % 64 == 4 -> conflict-free strided column reads)

// d_ws float layout: M[128*128] | u[128] | v[128] | s0[1]
#define WS_M  0
#define WS_U  (HID * HID)
#define WS_V  (WS_U + HID)
#define WS_S0 (WS_V + HID)

// ---------------------------------------------------------------------------
// Kernel 1: fold weights.  M = W_pep^T @ W_cdr3  (128x128, K=1024),
// u = W_pep^T b_cdr3, v = W_cdr3^T b_pep, s0 = b_pep . b_cdr3.
// 33 MFLOP total -> negligible; plain VALU.
// ---------------------------------------------------------------------------
__global__ __launch_bounds__(256) void mha_fold_weights(
    const float* __restrict__ Wc, const float* __restrict__ bc,
    const float* __restrict__ Wp, const float* __restrict__ bp,
    float* __restrict__ ws)
{
    int idx = blockIdx.x * 256 + threadIdx.x;      // 0..16383
    int i = idx >> 7;
    int j = idx & 127;
    float acc = 0.0f;
    for (int k = 0; k < 1024; ++k)
        acc = fmaf(Wp[k * HID + i], Wc[k * HID + j], acc);
    ws[WS_M + idx] = acc;

    if (idx < HID) {
        float ua = 0.0f, va = 0.0f;
        for (int k = 0; k < 1024; ++k) {
            ua = fmaf(Wp[k * HID + idx], bc[k], ua);
            va = fmaf(Wc[k * HID + idx], bp[k], va);
        }
        ws[WS_U + idx] = ua;
        ws[WS_V + idx] = va;
    }
    if (idx == HID) {
        float s = 0.0f;
        for (int k = 0; k < 1024; ++k) s = fmaf(bp[k], bc[k], s);
        ws[WS_S0] = s;
    }
}

// ---------------------------------------------------------------------------
// Kernel 2: one workgroup (8 wave32) per batch element.
//   T = P @ M          (64x128, fp32 WMMA, wave w owns T columns 16w..16w+15)
//   S = T @ C^T        (64x64,  fp32 WMMA, 2 tiles per wave) + bias terms, /1024
//   softmax over flattened 64x64, then
//   out[h] = sum_d P[d,h]*colsum[d] + sum_a C[a,h]*rowsum[a]  (all / Z)
// ---------------------------------------------------------------------------
__global__ __launch_bounds__(256) void mha_batch_kernel(
    const float* __restrict__ peptide, const float* __restrict__ cdr3,
    const float* __restrict__ ws, float* __restrict__ out)
{
    __shared__ float sP[SEQ * LDSP];
    __shared__ float sC[SEQ * LDSP];
    __shared__ float sT[SEQ * LDSP];
    __shared__ float sS[SEQ * SEQ];
    __shared__ float sPu[SEQ];
    __shared__ float sCv[SEQ];
    __shared__ float sRow[SEQ];
    __shared__ float sCol[SEQ];
    __shared__ float sRed[8];
    __shared__ float sScal[2];

    const int b    = blockIdx.x;
    const int tid  = threadIdx.x;
    const int lane = tid & 31;
    const int wv   = tid >> 5;          // 0..7
    const int m    = lane & 15;         // matrix row/col within fragment
    const int ks   = lane >> 4;         // K-half select (f32 WMMA A/B layout)

    const float* __restrict__ Mg = ws + WS_M;
    const float* __restrict__ Pg = peptide + (size_t)b * SEQ * HID;
    const float* __restrict__ Cg = cdr3    + (size_t)b * SEQ * HID;
    const float  s0 = ws[WS_S0];

    // Warm M (64 KB, shared by all 256 workgroups) into cache: global_prefetch_b8.
    __builtin_prefetch(&Mg[tid * 64], 0, 3);

    // ---- Stage P and C into LDS (padded rows, float4) --------------------
    for (int t = tid; t < SEQ * HID / 4; t += 256) {
        float4 vp = ((const float4*)Pg)[t];
        float4 vc = ((const float4*)Cg)[t];
        int row  = t >> 5;              // 32 float4 per 128-float row
        int col4 = (t & 31) << 2;
        *(float4*)&sP[row * LDSP + col4] = vp;
        *(float4*)&sC[row * LDSP + col4] = vc;
    }
    __syncthreads();

    // ---- Per-row bias dots: pu[p] = P[p,:].u  /  cv[c] = C[c,:].v --------
    if (tid < 128) {
        int r = tid & 63;
        const float* vec = (tid < 64) ? (ws + WS_U) : (ws + WS_V);
        const float* src = (tid < 64) ? &sP[r * LDSP] : &sC[r * LDSP];
        float a = 0.0f;
        for (int k = 0; k < HID; ++k) a = fmaf(src[k], vec[k], a);
        if (tid < 64) sPu[r] = a; else sCv[r] = a;
    }

    // ---- Phase 1: T = P @ M  (wave w -> T[:, 16w:16w+16], rows tiled x4) -
    const v8f vzero = {0.f, 0.f, 0.f, 0.f, 0.f, 0.f, 0.f, 0.f};
    v8f acc[4];
    for (int r = 0; r < 4; ++r) acc[r] = vzero;

    for (int k0 = 0; k0 < HID; k0 += 4) {
        const int kk = k0 + 2 * ks;
        v2f bfrag;                       // B[k,n] = M[kk + {0,1}, 16w + n]
        bfrag.x = Mg[kk * HID + 16 * wv + m];
        bfrag.y = Mg[(kk + 1) * HID + 16 * wv + m];
        #pragma unroll
        for (int r = 0; r < 4; ++r) {
            v2f afrag;                   // A[m,k] = P[16r+m, kk + {0,1}]
            const float* ap = &sP[(16 * r + m) * LDSP + kk];
            afrag.x = ap[0];
            afrag.y = ap[1];
            acc[r] = __builtin_amdgcn_wmma_f32_16x16x4_f32(
                false, afrag, false, bfrag, (short)0, acc[r], false, false);
        }
    }
    #pragma unroll
    for (int r = 0; r < 4; ++r)
        #pragma unroll
        for (int j = 0; j < 8; ++j)
            sT[(16 * r + j + 8 * ks) * LDSP + 16 * wv + m] = acc[r][j];
    __syncthreads();

    // ---- Phase 2: S = T @ C^T  (+bias terms, /1024) ----------------------
    for (int t = 2 * wv; t < 2 * wv + 2; ++t) {
        const int pi = t >> 2;           // pep tile row
        const int cj = t & 3;            // cdr3 tile col
        v8f sa = vzero;
        for (int k0 = 0; k0 < HID; k0 += 4) {
            const int kk = k0 + 2 * ks;
            v2f afrag, bfrag;
            const float* tp = &sT[(16 * pi + m) * LDSP + kk];
            afrag.x = tp[0];
            afrag.y = tp[1];
            const float* cp = &sC[(16 * cj + m) * LDSP + kk]; // B[k,n]=C[16cj+n,k]
            bfrag.x = cp[0];
            bfrag.y = cp[1];
            sa = __builtin_amdgcn_wmma_f32_16x16x4_f32(
                false, afrag, false, bfrag, (short)0, sa, false, false);
        }
        #pragma unroll
        for (int j = 0; j < 8; ++j) {
            const int p = 16 * pi + j + 8 * ks;
            const int c = 16 * cj + m;
            sS[p * SEQ + c] = (sa[j] + sPu[p] + sCv[c] + s0) * (1.0f / 1024.0f);
        }
    }
    __syncthreads();

    // ---- Phase 3: softmax over flattened 64x64 ---------------------------
    float lm = -3.402823466e38f;
    for (int t = tid; t < SEQ * SEQ; t += 256) lm = fmaxf(lm, sS[t]);
    for (int o = 16; o > 0; o >>= 1) lm = fmaxf(lm, __shfl_xor(lm, o));
    if (lane == 0) sRed[wv] = lm;
    __syncthreads();
    if (tid == 0) {
        float mx = sRed[0];
        for (int i = 1; i < 8; ++i) mx = fmaxf(mx, sRed[i]);
        sScal[0] = mx;
    }
    __syncthreads();
    const float mx = sScal[0];
    float lsum = 0.0f;
    for (int t = tid; t < SEQ * SEQ; t += 256) {
        float e = __expf(sS[t] - mx);
        sS[t] = e;
        lsum += e;
    }
    for (int o = 16; o > 0; o >>= 1) lsum += __shfl_xor(lsum, o);
    if (lane == 0) sRed[wv] = lsum;
    __syncthreads();
    if (tid == 0) {
        float z = 0.0f;
        for (int i = 0; i < 8; ++i) z += sRed[i];
        sScal[1] = z;
    }
    __syncthreads();
    const float invZ = 1.0f / sScal[1];

    // ---- Phase 4: row/col sums of exp(S) ---------------------------------
    if (tid < 64) {
        float r = 0.0f;
        for (int j = 0; j < SEQ; ++j) r += sS[tid * SEQ + j];
        sRow[tid] = r;                    // sum over cdr3 index -> pairs with C
    } else if (tid < 128) {
        const int c = tid - 64;
        float s = 0.0f;
        for (int i = 0; i < SEQ; ++i) s += sS[i * SEQ + c];
        sCol[c] = s;                      // sum over pep index -> pairs with P
    }
    __syncthreads();

    // ---- Phase 5: out[b,h] -----------------------------------------------
    if (tid < HID) {
        const int h = tid;
        float a = 0.0f;
        for (int i = 0; i < SEQ; ++i) {
            a = fmaf(sP[i * LDSP + h], sCol[i], a);
            a = fmaf(sC[i * LDSP + h], sRow[i], a);
        }
        out[(size_t)b * HID + h] = a * invZ;
    }
}

// ---------------------------------------------------------------------------
extern "C" void kernel_launch(void* const* d_in, const int* in_sizes, int n_in,
                              void* d_out, int out_size, void* d_ws, size_t ws_size,
                              hipStream_t stream) {
    (void)in_sizes; (void)n_in; (void)out_size; (void)ws_size;
    const float* peptide = (const float*)d_in[0];
    const float* cdr3    = (const float*)d_in[1];
    const float* W_cdr3  = (const float*)d_in[2];
    const float* b_cdr3  = (const float*)d_in[3];
    const float* W_pep   = (const float*)d_in[4];
    const float* b_pep   = (const float*)d_in[5];
    float*       out     = (float*)d_out;
    float*       ws      = (float*)d_ws;

    // Fold weights once: M = Wp^T Wc, u, v, s0  (16384 elements / 256 = 64 blocks)
    mha_fold_weights<<<64, 256, 0, stream>>>(W_cdr3, b_cdr3, W_pep, b_pep, ws);
    // Fused attention: one workgroup per batch element.
    mha_batch_kernel<<<NBATCH, 256, 0, stream>>>(peptide, cdr3, ws, out);
}